// SheafConvLayer_41592463295034
// MI455X (gfx1250) — compile-verified
//
#include <hip/hip_runtime.h>
#include <math.h>

typedef float v2f __attribute__((ext_vector_type(2)));
typedef float v8f __attribute__((ext_vector_type(8)));

#define FD 64          // feature dim
#define XROWS 128      // rows of x staged per block (8 waves * 16)
#define LSTR 68        // padded LDS row stride in floats (272B: 16B-aligned, bank-conflict-free)

// Async global->LDS 128-bit copy (GLOBAL_LOAD_ASYNC_TO_LDS_B128, ASYNCcnt-tracked).
__device__ __forceinline__ void async_copy_b128(void* lds_dst, const void* gsrc) {
  uint32_t loff = (uint32_t)(uintptr_t)lds_dst;            // LDS byte address
  uint64_t gaddr = (uint64_t)(uintptr_t)gsrc;
  asm volatile("global_load_async_to_lds_b128 %0, %1, off"
               :: "v"(loff), "v"(gaddr) : "memory");
}
__device__ __forceinline__ void async_wait0() {
  asm volatile("s_wait_asynccnt 0" ::: "memory");
}

// ---------------------------------------------------------------------------
// Kernel 1: per-node sheaf projections s1 = x.Wa, s2 = x.Wb ; zero diag accum
// ---------------------------------------------------------------------------
__global__ void k_node_proj(const float* __restrict__ x,
                            const float* __restrict__ Wsheaf,  // 128 floats
                            float* __restrict__ s1, float* __restrict__ s2,
                            float* __restrict__ diag, int N) {
  int n = blockIdx.x * blockDim.x + threadIdx.x;
  if (n >= N) return;
  const float4* xr = (const float4*)(x + (size_t)n * FD);
  const float4* wa = (const float4*)(Wsheaf);
  const float4* wb = (const float4*)(Wsheaf + FD);
  float a = 0.f, b = 0.f;
#pragma unroll
  for (int i = 0; i < FD / 4; ++i) {
    float4 xv = xr[i], av = wa[i], bv = wb[i];
    a += xv.x * av.x + xv.y * av.y + xv.z * av.z + xv.w * av.w;
    b += xv.x * bv.x + xv.y * bv.y + xv.z * bv.z + xv.w * bv.w;
  }
  s1[n] = a;
  s2[n] = b;
  diag[n] = 0.f;  // re-init every call (harness does not re-zero workspace)
}

// ---------------------------------------------------------------------------
// Kernel 2: maps[e] = tanh(s1[row]+s2[col]);  diag[row] += maps^2 (atomic)
// ---------------------------------------------------------------------------
__global__ void k_edge_maps(const int* __restrict__ row, const int* __restrict__ col,
                            const float* __restrict__ s1, const float* __restrict__ s2,
                            float* __restrict__ maps, float* __restrict__ diag, int E) {
  int e = blockIdx.x * blockDim.x + threadIdx.x;
  if (e >= E) return;
  int r = row[e], c = col[e];
  float m = tanhf(s1[r] + s2[c]);
  maps[e] = m;
  unsafeAtomicAdd(&diag[r], m * m);
}

// ---------------------------------------------------------------------------
// Kernel 3: dsi = rsqrt(diag+1); dcoef = dsi*diag*dsi  (= diag/(diag+1))
// ---------------------------------------------------------------------------
__global__ void k_dsi(const float* __restrict__ diag, float* __restrict__ dsi,
                      float* __restrict__ dcoef, int N) {
  int n = blockIdx.x * blockDim.x + threadIdx.x;
  if (n >= N) return;
  float d = diag[n];
  float r = rsqrtf(d + 1.0f);
  dsi[n] = r;
  dcoef[n] = r * d * r;
}

// ---------------------------------------------------------------------------
// Kernel 4: y = x @ W_lin.T + b_lin  via V_WMMA_F32_16X16X4_F32.
//  - W_lin (16KB) and this block's 128-row x tile are staged to LDS with
//    coalesced GLOBAL_LOAD_ASYNC_TO_LDS_B128 (ASYNCcnt), padded rows (LSTR=68)
//    so the half-wave ds_load_b64 fragment reads are bank-conflict-free.
//  - One wave computes a 16-row stripe: 4 N-tiles x 16 K-steps = 64 WMMAs.
//  - Epilogue writes y and acc = dcoef*y (diagonal Laplacian part); fast
//    unguarded path when the stripe is fully in range.
// A-frag (16x4 f32): lane m=lane&15 holds row m; K = 2*(lane>>4)+v.
// C/D: M = v + 8*(lane>>4), N = lane&15.
// ---------------------------------------------------------------------------
__global__ void __launch_bounds__(256) k_gemm_wmma(
    const float* __restrict__ x, const float* __restrict__ Wlin,
    const float* __restrict__ blin, const float* __restrict__ dcoef,
    float* __restrict__ y, float* __restrict__ acc, int N) {
  __shared__ __attribute__((aligned(16))) float Wlds[FD * LSTR];     // 17 KB
  __shared__ __attribute__((aligned(16))) float Xlds[XROWS * LSTR];  // 34 KB

  const int tid = threadIdx.x;
  const int lane = tid & 31;
  const int wave = tid >> 5;
  const int blockRow0 = blockIdx.x * XROWS;

  // ---- stage W_lin: 64 rows x 16 float4 chunks = 1024 chunks ----
#pragma unroll
  for (int i = 0; i < 4; ++i) {
    int chunk = tid + i * 256;
    int r = chunk >> 4, c4 = chunk & 15;
    async_copy_b128(&Wlds[r * LSTR + c4 * 4], Wlin + r * FD + c4 * 4);
  }
  // ---- stage x rows [blockRow0, blockRow0+128): 2048 chunks ----
#pragma unroll
  for (int i = 0; i < 8; ++i) {
    int chunk = tid + i * 256;
    int r = chunk >> 4, c4 = chunk & 15;
    int grow = blockRow0 + r;
    if (grow >= N) grow = N - 1;  // clamp (stores are guarded later)
    async_copy_b128(&Xlds[r * LSTR + c4 * 4], x + (size_t)grow * FD + c4 * 4);
  }
  async_wait0();
  __syncthreads();

  const int rowBase = blockRow0 + wave * 16;
  if (rowBase >= N) return;  // after barrier; wave-uniform so EXEC all-ones for WMMA
  const int m = lane & 15;
  const int g = lane >> 4;

  // A fragments from LDS (ds_load_b64, conflict-free)
  const float* xw = &Xlds[(wave * 16 + m) * LSTR];
  v2f afrag[16];
#pragma unroll
  for (int kk = 0; kk < 16; ++kk)
    afrag[kk] = *(const v2f*)(xw + 4 * kk + 2 * g);

  // Hoisted per-stripe diag coefficients
  const bool full = (rowBase + 16) <= N;
  float dc[8];
#pragma unroll
  for (int v = 0; v < 8; ++v) {
    int orow = rowBase + v + 8 * g;
    dc[v] = (orow < N) ? dcoef[orow] : 0.f;
  }

#pragma unroll
  for (int n0 = 0; n0 < FD; n0 += 16) {
    const int j = n0 + m;                      // output column owned by lane
    const float* wr = &Wlds[j * LSTR];         // column j of W_lin.T
    v8f c = {};
#pragma unroll
    for (int kk = 0; kk < 16; ++kk) {
      v2f b = *(const v2f*)(wr + 4 * kk + 2 * g);
      c = __builtin_amdgcn_wmma_f32_16x16x4_f32(
          /*neg_a=*/false, afrag[kk], /*neg_b=*/false, b,
          /*c_mod=*/(short)0, c, /*reuse_a=*/false, /*reuse_b=*/false);
    }
    const float bl = blin[j];
    const size_t base = (size_t)(rowBase + 8 * g) * FD + j;
    if (full) {
#pragma unroll
      for (int v = 0; v < 8; ++v) {
        const float yv = c[v] + bl;
        y[base + (size_t)v * FD] = yv;
        acc[base + (size_t)v * FD] = dc[v] * yv;  // diag part of Ly
      }
    } else {
#pragma unroll
      for (int v = 0; v < 8; ++v) {
        if (rowBase + v + 8 * g < N) {
          const float yv = c[v] + bl;
          y[base + (size_t)v * FD] = yv;
          acc[base + (size_t)v * FD] = dc[v] * yv;
        }
      }
    }
  }
}

// ---------------------------------------------------------------------------
// Kernel 5: per-edge scatter: acc[row] += dsi[row]*(-maps[e]*maps[rev])*dsi[col] * y[col]
// One edge per wave, 2 floats per lane, native f32 atomics into L2.
// ---------------------------------------------------------------------------
__global__ void k_scatter(const int* __restrict__ row, const int* __restrict__ col,
                          const int* __restrict__ ridx,
                          const float* __restrict__ maps, const float* __restrict__ dsi,
                          const float* __restrict__ y, float* __restrict__ acc, int E) {
  const int lane = threadIdx.x & 31;
  const int e = blockIdx.x * (blockDim.x >> 5) + (threadIdx.x >> 5);
  if (e >= E) return;
  if (lane == 0) {  // stream-ahead hints for the edge arrays (global_prefetch_b8)
    __builtin_prefetch(&row[e + 4096], 0, 1);
    __builtin_prefetch(&col[e + 4096], 0, 1);
    __builtin_prefetch(&ridx[e + 4096], 0, 1);
    __builtin_prefetch(&maps[e + 4096], 0, 1);
  }
  const int r = row[e], c = col[e];
  const float nm = -maps[e] * maps[ridx[e]] * dsi[r] * dsi[c];
  const float2 yv = *(const float2*)(y + (size_t)c * FD + 2 * lane);
  float* dst = acc + (size_t)r * FD + 2 * lane;
  unsafeAtomicAdd(dst + 0, nm * yv.x);
  unsafeAtomicAdd(dst + 1, nm * yv.y);
}

// ---------------------------------------------------------------------------
// Kernel 6: out = x - STEP * Ly  (STEP = 1.0), float4-vectorized
// ---------------------------------------------------------------------------
__global__ void k_out(const float* __restrict__ x, const float* __restrict__ acc,
                      float* __restrict__ out, int total4) {
  int i = blockIdx.x * blockDim.x + threadIdx.x;
  if (i >= total4) return;
  float4 xv = ((const float4*)x)[i];
  float4 av = ((const float4*)acc)[i];
  float4 o = {xv.x - av.x, xv.y - av.y, xv.z - av.z, xv.w - av.w};
  ((float4*)out)[i] = o;
}

// ---------------------------------------------------------------------------
extern "C" void kernel_launch(void* const* d_in, const int* in_sizes, int n_in,
                              void* d_out, int out_size, void* d_ws, size_t ws_size,
                              hipStream_t stream) {
  const float* x      = (const float*)d_in[0];
  const float* Wsheaf = (const float*)d_in[1];
  const float* Wlin   = (const float*)d_in[2];
  const float* blin   = (const float*)d_in[3];
  const int*   eidx   = (const int*)d_in[4];   // (2, E) flat
  // d_in[5] = left_idx (identity) — unused
  const int*   ridx   = (const int*)d_in[6];   // reverse-edge index

  const int N = in_sizes[0] / FD;
  const int E = in_sizes[4] / 2;
  const int* row = eidx;
  const int* col = eidx + E;

  // Workspace layout (floats)
  float* ws    = (float*)d_ws;
  float* s1    = ws;
  float* s2    = s1 + N;
  float* diag  = s2 + N;
  float* dsi   = diag + N;
  float* dcoef = dsi + N;
  float* maps  = dcoef + N;
  float* y     = maps + E;
  float* acc   = y + (size_t)N * FD;

  float* out = (float*)d_out;

  const int BT = 256;
  k_node_proj<<<(N + BT - 1) / BT, BT, 0, stream>>>(x, Wsheaf, s1, s2, diag, N);
  k_edge_maps<<<(E + BT - 1) / BT, BT, 0, stream>>>(row, col, s1, s2, maps, diag, E);
  k_dsi<<<(N + BT - 1) / BT, BT, 0, stream>>>(diag, dsi, dcoef, N);

  k_gemm_wmma<<<(N + XROWS - 1) / XROWS, BT, 0, stream>>>(
      x, Wlin, blin, dcoef, y, acc, N);

  const int wavesPerBlock = BT / 32;
  k_scatter<<<(E + wavesPerBlock - 1) / wavesPerBlock, BT, 0, stream>>>(
      row, col, ridx, maps, dsi, y, acc, E);

  const int total4 = N * FD / 4;
  k_out<<<(total4 + BT - 1) / BT, BT, 0, stream>>>(x, acc, out, total4);
}